// KGCN_1168231105082
// MI455X (gfx1250) — compile-verified
//
#include <hip/hip_runtime.h>
#include <math.h>

#define KFAN 16
#define DIM 64
#define NREL 32
#define WAVES 4
#define BLOCK (WAVES * 32)
#define WSTR 65   // padded strides -> conflict-free LDS
#define RSTR 65
#define XSTR 66

typedef float v2f __attribute__((ext_vector_type(2)));
typedef float v8f __attribute__((ext_vector_type(8)));

// Per-wave LDS handoff: DS ops of one wave execute in order and lanes are in
// lockstep (wave32 on SIMD32); we only need to stop compiler reordering.
#define WAVE_SYNC() __builtin_amdgcn_wave_barrier()

__device__ __forceinline__ float fsigm(float x) {
    return __builtin_amdgcn_rcpf(1.0f + __expf(-x));   // v_exp + v_rcp, no IEEE div
}

__global__ __launch_bounds__(BLOCK) void kgcn_fused(
    const int* __restrict__ uIdx, const int* __restrict__ vIdx,
    const int* __restrict__ adjEnt, const int* __restrict__ adjRel,
    const float* __restrict__ usrEmb, const float* __restrict__ relEmb,
    const float* __restrict__ entEmb, const float* __restrict__ Wm,
    const float* __restrict__ bias, float* __restrict__ outp, int B)
{
    __shared__ float sW[DIM * WSTR];          // W row-major, padded
    __shared__ float sRel[NREL * RSTR];       // rel_emb, padded
    __shared__ float sBias[DIM];
    __shared__ float sX[WAVES][16 * XSTR];    // per-wave 16x64 tile (X, then H1)
    __shared__ float sUser[WAVES][DIM];
    __shared__ float sVecBuf[WAVES][DIM];
    __shared__ float sUserRel[WAVES][NREL];
    __shared__ float sAttn0[WAVES][KFAN];
    __shared__ __align__(16) float sSc2[WAVES][KFAN][KFAN];   // hop2 scores -> exp
    __shared__ __align__(16) int   sIdx2[WAVES][KFAN][KFAN];  // hop2 entity ids
    __shared__ int   sE1[WAVES][KFAN];

    const int tid  = threadIdx.x;
    const int wave = tid >> 5;
    const int lane = tid & 31;
    const int d0 = 2 * lane, d1 = 2 * lane + 1;   // each lane owns 2 dims

    // Stage block-shared constants
    for (int i = tid; i < DIM * DIM; i += BLOCK)
        sW[(i >> 6) * WSTR + (i & 63)] = Wm[i];
    for (int i = tid; i < NREL * DIM; i += BLOCK)
        sRel[(i >> 6) * RSTR + (i & 63)] = relEmb[i];
    if (tid < DIM) sBias[tid] = bias[tid];

    int b = blockIdx.x * WAVES + wave;
    if (b >= B) b = B - 1;                 // tail waves redo last element
    const int ub = uIdx[b];
    const int vb = vIdx[b];

    const float2 uv = *(const float2*)(usrEmb + (size_t)ub * DIM + d0);
    sUser[wave][d0] = uv.x;
    sUser[wave][d1] = uv.y;
    __syncthreads();   // the ONLY block-wide barrier (covers sW/sRel/sBias too)

    // userRel[r] = user . rel_emb[r]  (NREL == 32 -> one per lane)
    {
        float acc = 0.0f;
        const float* rr = &sRel[lane * RSTR];
        #pragma unroll 8
        for (int d = 0; d < DIM; ++d) acc += sUser[wave][d] * rr[d];
        sUserRel[wave][lane] = acc;
    }
    if (lane < KFAN)
        sE1[wave][lane] = adjEnt[(size_t)vb * KFAN + lane];
    WAVE_SYNC();
    if (lane < KFAN) {
        const int r = adjRel[(size_t)vb * KFAN + lane];
        sAttn0[wave][lane] = sUserRel[wave][r];
    }
    WAVE_SYNC();
    // softmax -> attn0 (reused by BOTH aggregation iterations at hop 0)
    {
        float mx = -3.0e38f;
        #pragma unroll
        for (int k = 0; k < KFAN; ++k) mx = fmaxf(mx, sAttn0[wave][k]);
        if (lane < KFAN) sAttn0[wave][lane] = __expf(sAttn0[wave][lane] - mx);
        WAVE_SYNC();
        float s = 0.0f;
        #pragma unroll
        for (int k = 0; k < KFAN; ++k) s += sAttn0[wave][k];
        const float inv = __builtin_amdgcn_rcpf(s);
        if (lane < KFAN) sAttn0[wave][lane] *= inv;
        WAVE_SYNC();
    }

    // Prefetch ALL 256 hop-2 indices + scores (coalesced, 2 rows per pass)
    #pragma unroll
    for (int mm = 0; mm < 8; ++mm) {
        const int m = 2 * mm + (lane >> 4);
        const int k = lane & 15;
        const int em = sE1[wave][m];
        sIdx2[wave][m][k] = adjEnt[(size_t)em * KFAN + k];
        const int r = adjRel[(size_t)em * KFAN + k];
        sSc2[wave][m][k] = sUserRel[wave][r];
    }
    WAVE_SYNC();

    // ---- iter0 hop1: X[m] = ent_emb[e1[m]] + softmax-weighted hop2 mix
    //      (fused: x0 accumulation for iter0 hop0 rides along on the self rows)
    const float2 e0v = *(const float2*)(entEmb + (size_t)vb * DIM + d0);
    float x0a = e0v.x, x0b = e0v.y;
    for (int m = 0; m < KFAN; ++m) {
        // row of 16 scores via 4 uniform b128 broadcasts
        const float4 s0 = *(const float4*)&sSc2[wave][m][0];
        const float4 s1 = *(const float4*)&sSc2[wave][m][4];
        const float4 s2 = *(const float4*)&sSc2[wave][m][8];
        const float4 s3 = *(const float4*)&sSc2[wave][m][12];
        const float mx =
            fmaxf(fmaxf(fmaxf(fmaxf(s0.x, s0.y), fmaxf(s0.z, s0.w)),
                        fmaxf(fmaxf(s1.x, s1.y), fmaxf(s1.z, s1.w))),
                  fmaxf(fmaxf(fmaxf(s2.x, s2.y), fmaxf(s2.z, s2.w)),
                        fmaxf(fmaxf(s3.x, s3.y), fmaxf(s3.z, s3.w))));
        if (lane < KFAN)
            sSc2[wave][m][lane] = __expf(sSc2[wave][m][lane] - mx);
        WAVE_SYNC();
        const float4 e0_ = *(const float4*)&sSc2[wave][m][0];
        const float4 e1_ = *(const float4*)&sSc2[wave][m][4];
        const float4 e2_ = *(const float4*)&sSc2[wave][m][8];
        const float4 e3_ = *(const float4*)&sSc2[wave][m][12];
        const float wv[16] = {e0_.x, e0_.y, e0_.z, e0_.w,
                              e1_.x, e1_.y, e1_.z, e1_.w,
                              e2_.x, e2_.y, e2_.z, e2_.w,
                              e3_.x, e3_.y, e3_.z, e3_.w};
        float s = 0.0f;
        #pragma unroll
        for (int k = 0; k < KFAN; ++k) s += wv[k];
        const float inv = __builtin_amdgcn_rcpf(s);

        const int4 i0 = *(const int4*)&sIdx2[wave][m][0];
        const int4 i1 = *(const int4*)&sIdx2[wave][m][4];
        const int4 i2 = *(const int4*)&sIdx2[wave][m][8];
        const int4 i3 = *(const int4*)&sIdx2[wave][m][12];
        const int ix[16] = {i0.x, i0.y, i0.z, i0.w, i1.x, i1.y, i1.z, i1.w,
                            i2.x, i2.y, i2.z, i2.w, i3.x, i3.y, i3.z, i3.w};
        float a0 = 0.0f, a1 = 0.0f;
        #pragma unroll
        for (int k = 0; k < KFAN; ++k) {      // 16 independent 256B row gathers
            const float2 ev = *(const float2*)(entEmb + (size_t)ix[k] * DIM + d0);
            a0 += wv[k] * ev.x;
            a1 += wv[k] * ev.y;
        }
        const int em = sE1[wave][m];
        const float2 sv = *(const float2*)(entEmb + (size_t)em * DIM + d0);
        const float w0 = sAttn0[wave][m];
        x0a += w0 * sv.x;
        x0b += w0 * sv.y;
        sX[wave][m * XSTR + d0] = sv.x + a0 * inv;
        sX[wave][m * XSTR + d1] = sv.y + a1 * inv;
    }
    WAVE_SYNC();

    // ---- H1 = sigmoid(X @ W + b) via f32 WMMA 16x16x4, 16 K-steps x 4 N-tiles
    const int arow  = lane & 15;
    const int khalf = (lane >> 4) << 1;       // lanes 0-15: K 0,1 ; 16-31: K 2,3
    const int ncol  = lane & 15;
    const float* xrow = &sX[wave][arow * XSTR];

    v8f acc[4] = {};
    #pragma unroll 4
    for (int kk = 0; kk < 16; ++kk) {
        v2f a;
        a.x = xrow[4 * kk + khalf];
        a.y = xrow[4 * kk + khalf + 1];
        const float* wr0 = &sW[(4 * kk + khalf) * WSTR];
        const float* wr1 = wr0 + WSTR;
        #pragma unroll
        for (int nt = 0; nt < 4; ++nt) {
            v2f bb;
            bb.x = wr0[ncol + nt * 16];
            bb.y = wr1[ncol + nt * 16];
            acc[nt] = __builtin_amdgcn_wmma_f32_16x16x4_f32(
                false, a, false, bb, (short)0, acc[nt], false, false);
        }
    }
    WAVE_SYNC();
    {   // D layout: VGPR i -> M = i (lanes 0-15) / i+8 (lanes 16-31), N = nt*16+ncol
        const int mbase = (lane >> 4) << 3;
        #pragma unroll
        for (int nt = 0; nt < 4; ++nt) {
            const int   n  = nt * 16 + ncol;
            const float bn = sBias[n];
            #pragma unroll
            for (int i = 0; i < 8; ++i)
                sX[wave][(mbase + i) * XSTR + n] = fsigm(acc[nt][i] + bn);
        }
    }
    WAVE_SYNC();

    // ---- iter0 hop0: h0 = sigmoid(x0 @ W + b)
    sVecBuf[wave][d0] = x0a;
    sVecBuf[wave][d1] = x0b;
    WAVE_SYNC();
    float h0a = sBias[d0], h0b = sBias[d1];
    #pragma unroll 8
    for (int dd = 0; dd < DIM; ++dd) {
        const float xv = sVecBuf[wave][dd];
        h0a += xv * sW[dd * WSTR + d0];
        h0b += xv * sW[dd * WSTR + d1];
    }
    h0a = fsigm(h0a);
    h0b = fsigm(h0b);

    // ---- iter1 hop0: item = tanh((h0 + attn0 . H1) @ W + b)
    float g0 = 0.0f, g1 = 0.0f;
    #pragma unroll
    for (int k = 0; k < KFAN; ++k) {
        const float w = sAttn0[wave][k];
        g0 += w * sX[wave][k * XSTR + d0];
        g1 += w * sX[wave][k * XSTR + d1];
    }
    WAVE_SYNC();
    sVecBuf[wave][d0] = h0a + g0;
    sVecBuf[wave][d1] = h0b + g1;
    WAVE_SYNC();
    float it0 = sBias[d0], it1 = sBias[d1];
    #pragma unroll 8
    for (int dd = 0; dd < DIM; ++dd) {
        const float xv = sVecBuf[wave][dd];
        it0 += xv * sW[dd * WSTR + d0];
        it1 += xv * sW[dd * WSTR + d1];
    }
    it0 = tanhf(it0);
    it1 = tanhf(it1);

    // ---- out[b] = sigmoid(user . item), wave32 reduction
    float part = uv.x * it0 + uv.y * it1;
    #pragma unroll
    for (int off = 16; off >= 1; off >>= 1)
        part += __shfl_xor(part, off, 32);
    if (lane == 0) outp[b] = fsigm(part);
}

extern "C" void kernel_launch(void* const* d_in, const int* in_sizes, int n_in,
                              void* d_out, int out_size, void* d_ws, size_t ws_size,
                              hipStream_t stream) {
    const int*   u       = (const int*)d_in[0];
    const int*   v       = (const int*)d_in[1];
    const int*   adj_ent = (const int*)d_in[2];
    const int*   adj_rel = (const int*)d_in[3];
    const float* usr_emb = (const float*)d_in[4];
    const float* rel_emb = (const float*)d_in[5];
    const float* ent_emb = (const float*)d_in[6];
    const float* W       = (const float*)d_in[7];
    const float* b       = (const float*)d_in[8];
    float* out = (float*)d_out;
    const int B = in_sizes[0];
    const int grid = (B + WAVES - 1) / WAVES;
    hipLaunchKernelGGL(kgcn_fused, dim3(grid), dim3(BLOCK), 0, stream,
                       u, v, adj_ent, adj_rel, usr_emb, rel_emb, ent_emb, W, b, out, B);
}